// HyperCrossAttention_47622597378302
// MI455X (gfx1250) — compile-verified
//
#include <hip/hip_runtime.h>
#include <hip/hip_bf16.h>
#include <math.h>

typedef __attribute__((ext_vector_type(2))) float v2f;
typedef __attribute__((ext_vector_type(4))) float v4f;
typedef __attribute__((ext_vector_type(8))) float v8f;

#define N_SEQ   8192
#define DHEAD   64
#define NPROJ   8
#define BLOCK_K 128
#define SAMPLE_K 128
#define NBLK    (N_SEQ / BLOCK_K)   // 64

#define KPAD 68   // ldsK row stride (floats): bank = 4n+c, conflict-free
#define VPAD 68   // ldsV row stride (floats)
#define PPAD 20   // per-wave P scratch row stride (banks 20m distinct mod 64)

__device__ __forceinline__ v8f wmma_f32(v2f a, v2f b, v8f c) {
  // V_WMMA_F32_16X16X4_F32 : D = A(16x4 f32) x B(4x16 f32) + C(16x16 f32)
  return __builtin_amdgcn_wmma_f32_16x16x4_f32(false, a, false, b, (short)0, c,
                                               false, false);
}

// Async 16B copy global -> LDS (CDNA5 GLOBAL_LOAD_ASYNC_TO_LDS_B128, ASYNCcnt)
#define ASYNC_CP16(ldsOff, gaddr, OFF)                                         \
  asm volatile("global_load_async_to_lds_b128 %0, %1, off offset:" #OFF        \
               ::"v"(ldsOff), "v"(gaddr) : "memory")

// ---------------------------------------------------------------------------
// Kernel 1: LSH hash of q rows and k rows.  GRAY[x] == x ^ (x>>1).
// ---------------------------------------------------------------------------
__global__ __launch_bounds__(256)
void hash_kernel(const float* __restrict__ q, const float* __restrict__ kv,
                 const float* __restrict__ proj, unsigned int* __restrict__ hq,
                 unsigned int* __restrict__ hk, int BH) {
  __shared__ float pr[DHEAD * NPROJ];
  for (int i = threadIdx.x; i < DHEAD * NPROJ; i += blockDim.x) pr[i] = proj[i];
  __syncthreads();

  long long gid = (long long)blockIdx.x * blockDim.x + threadIdx.x;
  long long total = 2LL * BH * N_SEQ;
  if (gid >= total) return;
  int which = (int)(gid / ((long long)BH * N_SEQ));
  int rem   = (int)(gid % ((long long)BH * N_SEQ));
  int bh = rem / N_SEQ, n = rem % N_SEQ;

  const float* row = (which == 0)
      ? q  + ((long long)bh * N_SEQ + n) * DHEAD
      : kv + (((long long)bh * 2 + 0) * N_SEQ + n) * DHEAD;

  float acc[NPROJ];
#pragma unroll
  for (int p = 0; p < NPROJ; ++p) acc[p] = 0.f;
#pragma unroll
  for (int c = 0; c < DHEAD / 4; ++c) {
    v4f x = *(const v4f*)(row + 4 * c);
#pragma unroll
    for (int e = 0; e < 4; ++e) {
      float xv = x[e];
      const float* pp = pr + (4 * c + e) * NPROJ;
#pragma unroll
      for (int p = 0; p < NPROJ; ++p) acc[p] = fmaf(xv, pp[p], acc[p]);
    }
  }
  unsigned int code = 0;
#pragma unroll
  for (int p = 0; p < NPROJ; ++p) code |= (acc[p] > 0.f ? 1u : 0u) << p;
  unsigned int g = code ^ (code >> 1);
  (which == 0 ? hq : hk)[(long long)bh * N_SEQ + n] = g;
}

// ---------------------------------------------------------------------------
// Kernel 2: stable argsort of 8-bit hash per (b,h) — 2-pass 4-bit LSD radix,
// fully in LDS.  One 256-thread WG per (which, bh).
// ---------------------------------------------------------------------------
__global__ __launch_bounds__(256)
void sort_kernel(const unsigned int* __restrict__ hq,
                 const unsigned int* __restrict__ hk,
                 int* __restrict__ idxq, int* __restrict__ idxk, int BH) {
  __shared__ unsigned char  keys[N_SEQ];
  __shared__ unsigned short idxA[N_SEQ];
  __shared__ unsigned short idxB[N_SEQ];
  __shared__ unsigned int   hist[16][256];
  __shared__ unsigned int   digitBase[16];

  int bid = blockIdx.x;
  int which = bid / BH, bh = bid % BH;
  const unsigned int* h = (which == 0 ? hq : hk) + (long long)bh * N_SEQ;
  int tid = threadIdx.x;

  for (int i = tid; i < N_SEQ; i += 256) {
    keys[i] = (unsigned char)h[i];
    idxA[i] = (unsigned short)i;
  }
  __syncthreads();

  for (int round = 0; round < 2; ++round) {
    int shift = round * 4;
    unsigned short* src = (round == 0) ? idxA : idxB;
    unsigned short* dst = (round == 0) ? idxB : idxA;

    for (int i = tid; i < 16 * 256; i += 256) (&hist[0][0])[i] = 0u;
    __syncthreads();

    const int CH = N_SEQ / 256;   // 32 contiguous elements per thread (stable)
    int base = tid * CH;
    for (int e = 0; e < CH; ++e) {
      int d = (keys[src[base + e]] >> shift) & 15;
      hist[d][tid]++;
    }
    __syncthreads();

    if (tid < 16) {               // exclusive scan over chunks, per digit
      unsigned int run = 0;
      for (int t = 0; t < 256; ++t) {
        unsigned int v = hist[tid][t]; hist[tid][t] = run; run += v;
      }
      digitBase[tid] = run;
    }
    __syncthreads();
    if (tid == 0) {               // exclusive scan over digit totals
      unsigned int run = 0;
      for (int d = 0; d < 16; ++d) {
        unsigned int v = digitBase[d]; digitBase[d] = run; run += v;
      }
    }
    __syncthreads();

    for (int e = 0; e < CH; ++e) {  // stable scatter
      unsigned short j = src[base + e];
      int d = (keys[j] >> shift) & 15;
      unsigned int pos = digitBase[d] + hist[d][tid];
      hist[d][tid] = hist[d][tid] + 1;
      dst[pos] = j;
    }
    __syncthreads();
  }

  int* out = (which == 0 ? idxq : idxk) + (long long)bh * N_SEQ;
  for (int i = tid; i < N_SEQ; i += 256) out[i] = (int)idxA[i];
}

// ---------------------------------------------------------------------------
// Kernel 3: fused block + sampled attention with combined softmax, scattered
// output.  One 256-thread WG (8 waves) per (b,h,block): 128 q x 256 k x 64 d.
// Sampled columns carry bias log(64) (or -FLT_MAX when in q's own block).
// ---------------------------------------------------------------------------
__global__ __launch_bounds__(256)
void attn_kernel(const float* __restrict__ q, const float* __restrict__ kv,
                 const int* __restrict__ sampled,
                 const int* __restrict__ idxq, const int* __restrict__ idxk,
                 float* __restrict__ out, int BH) {
  __shared__ float ldsK[256 * KPAD];            // 69632 B
  __shared__ float ldsV[256 * VPAD];            // 69632 B
  __shared__ float ldsP[8 * 2 * 16 * PPAD];     // 20480 B (per-wave, 2 bufs)
  __shared__ float ldsBias[256];                //  1024 B

  int gid = blockIdx.x;
  int c  = gid % NBLK;
  int bh = gid / NBLK;
  int tid = threadIdx.x;

  const int* qIdx = idxq + (long long)bh * N_SEQ;
  const int* kIdx = idxk + (long long)bh * N_SEQ;
  const float* kBase = kv + ((long long)bh * 2 + 0) * N_SEQ * DHEAD;
  const float* vBase = kv + ((long long)bh * 2 + 1) * N_SEQ * DHEAD;

  // ---- async-gather K/V rows into LDS: j<128 block part, j>=128 sampled ----
  {
    int j = tid;
    int srow; float bias;
    if (j < BLOCK_K) {
      srow = c * BLOCK_K + j; bias = 0.f;
    } else {
      int s = sampled[(long long)bh * SAMPLE_K + (j - BLOCK_K)];
      srow = s;
      bias = 4.1588830833596715f;                 // log(nk/SAMPLE) = log(64)
      if ((s / BLOCK_K) == c) bias = -3.402823466e38f;  // finfo(f32).min
    }
    ldsBias[j] = bias;
    int kr = kIdx[srow];
    unsigned long long ga_k =
        (unsigned long long)(const void*)(kBase + (long long)kr * DHEAD);
    unsigned long long ga_v =
        (unsigned long long)(const void*)(vBase + (long long)kr * DHEAD);
    unsigned int lk = (unsigned int)(unsigned long long)(void*)&ldsK[j * KPAD];
    unsigned int lv = (unsigned int)(unsigned long long)(void*)&ldsV[j * VPAD];
    ASYNC_CP16(lk, ga_k, 0);   ASYNC_CP16(lv, ga_v, 0);
    ASYNC_CP16(lk, ga_k, 16);  ASYNC_CP16(lv, ga_v, 16);
    ASYNC_CP16(lk, ga_k, 32);  ASYNC_CP16(lv, ga_v, 32);
    ASYNC_CP16(lk, ga_k, 48);  ASYNC_CP16(lv, ga_v, 48);
    ASYNC_CP16(lk, ga_k, 64);  ASYNC_CP16(lv, ga_v, 64);
    ASYNC_CP16(lk, ga_k, 80);  ASYNC_CP16(lv, ga_v, 80);
    ASYNC_CP16(lk, ga_k, 96);  ASYNC_CP16(lv, ga_v, 96);
    ASYNC_CP16(lk, ga_k, 112); ASYNC_CP16(lv, ga_v, 112);
    ASYNC_CP16(lk, ga_k, 128); ASYNC_CP16(lv, ga_v, 128);
    ASYNC_CP16(lk, ga_k, 144); ASYNC_CP16(lv, ga_v, 144);
    ASYNC_CP16(lk, ga_k, 160); ASYNC_CP16(lv, ga_v, 160);
    ASYNC_CP16(lk, ga_k, 176); ASYNC_CP16(lv, ga_v, 176);
    ASYNC_CP16(lk, ga_k, 192); ASYNC_CP16(lv, ga_v, 192);
    ASYNC_CP16(lk, ga_k, 208); ASYNC_CP16(lv, ga_v, 208);
    ASYNC_CP16(lk, ga_k, 224); ASYNC_CP16(lv, ga_v, 224);
    ASYNC_CP16(lk, ga_k, 240); ASYNC_CP16(lv, ga_v, 240);
  }
  asm volatile("s_wait_asynccnt 0x0" ::: "memory");
  __syncthreads();

  int w = tid >> 5, lane = tid & 31;
  int lh = lane >> 4, ln = lane & 15;

  // ---- Q A-fragments for this wave's 16 rows (lane l: [l%16][4c+2h+i]) ----
  v2f qf[16];
  {
    int qr = qIdx[c * BLOCK_K + w * 16 + ln];
    const float* qp = q + ((long long)bh * N_SEQ + qr) * DHEAD + 2 * lh;
#pragma unroll
    for (int c2 = 0; c2 < 16; ++c2) qf[c2] = *(const v2f*)(qp + 4 * c2);
  }

  const float scale = 0.125f;   // 64^-0.5
  const v8f Z = {0.f, 0.f, 0.f, 0.f, 0.f, 0.f, 0.f, 0.f};

  // ---- S = Q K^T : 16 tiles of 16x16 over 256 columns ----
  float sbuf[16][8];
#pragma unroll
  for (int t = 0; t < 16; ++t) {
    const float* kb = &ldsK[(t * 16 + ln) * KPAD + 2 * lh];
    v2f kf[16];                       // batch all B-fragments: loads pipeline
#pragma unroll
    for (int c2 = 0; c2 < 16; ++c2) kf[c2] = *(const v2f*)(kb + 4 * c2);
    v8f acc0 = Z, acc1 = Z;           // two independent WMMA chains
#pragma unroll
    for (int c2 = 0; c2 < 8; ++c2) {
      acc0 = wmma_f32(qf[2 * c2 + 0], kf[2 * c2 + 0], acc0);
      acc1 = wmma_f32(qf[2 * c2 + 1], kf[2 * c2 + 1], acc1);
    }
    float bias = ldsBias[t * 16 + ln];
#pragma unroll
    for (int r = 0; r < 8; ++r)
      sbuf[t][r] = (acc0[r] + acc1[r]) * scale + bias;
  }

  // ---- softmax over 256 columns; row m = r + 8*lh ----
  float rmax[8], rsum[8];
#pragma unroll
  for (int r = 0; r < 8; ++r) {
    float m = sbuf[0][r];
#pragma unroll
    for (int t = 1; t < 16; ++t) m = fmaxf(m, sbuf[t][r]);
#pragma unroll
    for (int off = 8; off >= 1; off >>= 1) m = fmaxf(m, __shfl_xor(m, off, 32));
    rmax[r] = m;
  }
#pragma unroll
  for (int r = 0; r < 8; ++r) {
    float s = 0.f;
#pragma unroll
    for (int t = 0; t < 16; ++t) {
      float p = __expf(sbuf[t][r] - rmax[r]);
      sbuf[t][r] = p;
      s += p;
    }
#pragma unroll
    for (int off = 8; off >= 1; off >>= 1) s += __shfl_xor(s, off, 32);
    rsum[r] = s;
  }

  // ---- O = P V : double-buffered per-wave P staging through LDS ----
  v8f dacc[4];
#pragma unroll
  for (int dt = 0; dt < 4; ++dt) dacc[dt] = Z;

  float* pw = &ldsP[w * 2 * 16 * PPAD];
  // pre-stage tile 0 into buffer 0
#pragma unroll
  for (int r = 0; r < 8; ++r) pw[(r + 8 * lh) * PPAD + ln] = sbuf[0][r];

#pragma unroll
  for (int kt = 0; kt < 16; ++kt) {
    if (kt < 15) {                      // stage next tile into other buffer
      float* pn = pw + ((kt + 1) & 1) * 16 * PPAD;
#pragma unroll
      for (int r = 0; r < 8; ++r)
        pn[(r + 8 * lh) * PPAD + ln] = sbuf[kt + 1][r];
    }
    float* pb = pw + (kt & 1) * 16 * PPAD;
    v2f af[4];
#pragma unroll
    for (int c3 = 0; c3 < 4; ++c3)
      af[c3] = *(const v2f*)(pb + ln * PPAD + 4 * c3 + 2 * lh);
    v2f vf[16];                         // batch all V B-fragments
#pragma unroll
    for (int dt = 0; dt < 4; ++dt) {
#pragma unroll
      for (int c3 = 0; c3 < 4; ++c3) {
        int kr = kt * 16 + 4 * c3 + 2 * lh;
        v2f b2;
        b2.x = ldsV[(kr + 0) * VPAD + dt * 16 + ln];
        b2.y = ldsV[(kr + 1) * VPAD + dt * 16 + ln];
        vf[dt * 4 + c3] = b2;
      }
    }
#pragma unroll
    for (int dt = 0; dt < 4; ++dt) {    // 4 independent accumulator chains
#pragma unroll
      for (int c3 = 0; c3 < 4; ++c3)
        dacc[dt] = wmma_f32(af[c3], vf[dt * 4 + c3], dacc[dt]);
    }
  }

  // ---- normalize and scatter by q_idx (== gather by q_idx_inv) ----
#pragma unroll
  for (int r = 0; r < 8; ++r) {
    float inv = 1.0f / rsum[r];
    int m = r + 8 * lh;
    int qr = qIdx[c * BLOCK_K + w * 16 + m];
    float* ob = out + ((long long)bh * N_SEQ + qr) * DHEAD;
#pragma unroll
    for (int dt = 0; dt < 4; ++dt) ob[dt * 16 + ln] = dacc[dt][r] * inv;
  }
}

// ---------------------------------------------------------------------------
extern "C" void kernel_launch(void* const* d_in, const int* in_sizes, int n_in,
                              void* d_out, int out_size, void* d_ws,
                              size_t ws_size, hipStream_t stream) {
  const float* q       = (const float*)d_in[0];
  const float* kv      = (const float*)d_in[1];
  const float* proj    = (const float*)d_in[2];
  const int*   sampled = (const int*)d_in[3];
  float* out = (float*)d_out;

  int BH = in_sizes[0] / (N_SEQ * DHEAD);   // B*H = 32

  unsigned int* hq = (unsigned int*)d_ws;
  unsigned int* hk = hq + (size_t)BH * N_SEQ;
  int* idxq = (int*)(hk + (size_t)BH * N_SEQ);
  int* idxk = idxq + (size_t)BH * N_SEQ;

  {
    long long total = 2LL * BH * N_SEQ;
    int blocks = (int)((total + 255) / 256);
    hipLaunchKernelGGL(hash_kernel, dim3(blocks), dim3(256), 0, stream,
                       q, kv, proj, hq, hk, BH);
  }
  hipLaunchKernelGGL(sort_kernel, dim3(2 * BH), dim3(256), 0, stream,
                     hq, hk, idxq, idxk, BH);
  hipLaunchKernelGGL(attn_kernel, dim3(BH * NBLK), dim3(256), 0, stream,
                     q, kv, sampled, idxq, idxk, out, BH);
}